// mLSTMCell_45724221833938
// MI455X (gfx1250) — compile-verified
//
#include <hip/hip_runtime.h>
#include <math.h>

#define Bsz 64
#define Hd  1024
#define BH  (Bsz * Hd)

typedef float v2f __attribute__((ext_vector_type(2)));
typedef float v4f __attribute__((ext_vector_type(4)));
typedef float v8f __attribute__((ext_vector_type(8)));

struct GemmArgs {
    const float* W[6];
    const float* bias[6];
};

// ---------------------------------------------------------------------------
// Kernel 1: six gate linears  lin[g] = x @ W_g^T + b_g   via V_WMMA_F32_16X16X4_F32
// One wave handles one (gate, 16-column tile) and all 64 rows (4 M-tiles,
// 4 accumulators) so the W tile is loaded once per K-step and reused 4x.
// ---------------------------------------------------------------------------
__global__ void gemm_gates_kernel(const float* __restrict__ x, GemmArgs ga,
                                  float* __restrict__ lin)
{
    const int wid  = threadIdx.x >> 5;        // wave in block (0..3)
    const int lane = threadIdx.x & 31;
    const int g    = blockIdx.y;              // gate 0..5
    const int colTile = blockIdx.x * 4 + wid; // 0..63
    const int mrow = lane & 15;
    const int half = lane >> 4;               // 0: K=0,1 ; 1: K=2,3
    const int n_col = colTile * 16 + mrow;    // this lane's output column

    const v2f* __restrict__ wrow = (const v2f*)(ga.W[g] + (size_t)n_col * Hd);
    const v2f* __restrict__ x2   = (const v2f*)x;

    v8f acc0 = {}, acc1 = {}, acc2 = {}, acc3 = {};

    for (int k0 = 0; k0 < Hd; k0 += 4) {
        const int kb = (k0 >> 1) + half;      // v2f index of (k0 + 2*half)

        v2f bvec = wrow[kb];                  // B[k][n] = W[n][k]
        v2f a0 = x2[(size_t)( 0 + mrow) * (Hd / 2) + kb];
        v2f a1 = x2[(size_t)(16 + mrow) * (Hd / 2) + kb];
        v2f a2 = x2[(size_t)(32 + mrow) * (Hd / 2) + kb];
        v2f a3 = x2[(size_t)(48 + mrow) * (Hd / 2) + kb];

        acc0 = __builtin_amdgcn_wmma_f32_16x16x4_f32(false, a0, false, bvec,
                                                     (short)0, acc0, false, false);
        acc1 = __builtin_amdgcn_wmma_f32_16x16x4_f32(false, a1, false, bvec,
                                                     (short)0, acc1, false, false);
        acc2 = __builtin_amdgcn_wmma_f32_16x16x4_f32(false, a2, false, bvec,
                                                     (short)0, acc2, false, false);
        acc3 = __builtin_amdgcn_wmma_f32_16x16x4_f32(false, a3, false, bvec,
                                                     (short)0, acc3, false, false);
    }

    const float bias = ga.bias[g][n_col];
    float* __restrict__ out = lin + (size_t)g * BH;
#pragma unroll
    for (int r = 0; r < 8; ++r) {
        const int m = r + 8 * half;           // C/D layout: lanes>=16 hold M=8..15
        out[(size_t)( 0 + m) * Hd + n_col] = acc0[r] + bias;
        out[(size_t)(16 + m) * Hd + n_col] = acc1[r] + bias;
        out[(size_t)(32 + m) * Hd + n_col] = acc2[r] + bias;
        out[(size_t)(48 + m) * Hd + n_col] = acc3[r] + bias;
    }
}

// ---------------------------------------------------------------------------
// Kernel 2: pointwise gates + per-batch reduction for the divisor.
// One block (256 threads) per batch element b.
// ---------------------------------------------------------------------------
__global__ void gates_kernel(const float* __restrict__ lin,
                             const float* __restrict__ n_in,
                             const float* __restrict__ m_in,
                             float* __restrict__ f_t, float* __restrict__ o_t,
                             float* __restrict__ ipv, float* __restrict__ ksc,
                             float* __restrict__ divv,
                             float* __restrict__ n_out, float* __restrict__ m_out)
{
    const int b   = blockIdx.x;
    const int tid = threadIdx.x;
    __shared__ float red[8];

    float local = 0.0f;
    for (int h = tid; h < Hd; h += 256) {
        const int idx = b * Hd + h;
        const float it     = lin[0 * BH + idx];
        const float ft_pre = lin[1 * BH + idx];
        const float ot_pre = lin[2 * BH + idx];
        const float qv     = lin[3 * BH + idx];
        const float kv     = lin[4 * BH + idx] * 0.03125f; // 1/sqrt(1024)
        const float vv     = lin[5 * BH + idx];

        const float f_sig  = 1.0f / (1.0f + expf(-ft_pre));
        const float o_sig  = 1.0f / (1.0f + expf(-ot_pre));
        // stable log(sigmoid(x)) = min(x,0) - log1p(exp(-|x|))
        const float logsig = fminf(ft_pre, 0.0f) - log1pf(expf(-fabsf(ft_pre)));
        const float mt     = fmaxf(logsig + m_in[idx], it);
        const float ipr    = expf(it - mt);
        const float nt     = f_sig * n_in[idx] + ipr * kv;

        f_t[idx] = f_sig;
        o_t[idx] = o_sig;
        ipv[idx] = ipr * vv;
        ksc[idx] = kv;
        n_out[idx] = nt;
        m_out[idx] = mt;
        local += nt * qv;
    }

    // wave32 reduce, then cross-wave via LDS
#pragma unroll
    for (int off = 16; off > 0; off >>= 1)
        local += __shfl_down(local, off, 32);
    const int wv = tid >> 5, ln = tid & 31;
    if (ln == 0) red[wv] = local;
    __syncthreads();
    if (tid == 0) {
        float s = 0.0f;
#pragma unroll
        for (int w = 0; w < 8; ++w) s += red[w];
        divv[b] = fmaxf(fabsf(s), 1.0f);
    }
}

// ---------------------------------------------------------------------------
// Kernel 3: the 512 MB stream. One wave per C-row (65536 rows):
//   C_t[row,:] = f*C[row,:] + (i'*v)*k[:]   (non-temporal: pure stream)
//   h[row]     = o * (C_t[row,:] . q) / divisor[b]   (fused, no C_t re-read)
// ---------------------------------------------------------------------------
__global__ void stream_kernel(const float* __restrict__ C,
                              const float* __restrict__ f_t,
                              const float* __restrict__ o_t,
                              const float* __restrict__ ipv,
                              const float* __restrict__ ksc,
                              const float* __restrict__ qlin,
                              const float* __restrict__ divv,
                              float* __restrict__ Cout,
                              float* __restrict__ hout)
{
    const int wv   = threadIdx.x >> 5;
    const int lane = threadIdx.x & 31;
    const int row  = blockIdx.x * 8 + wv;     // 0 .. 65535
    const int b    = row >> 10;

    const float fs = f_t[row];
    const float cs = ipv[row];

    const v4f* __restrict__ crow = (const v4f*)(C    + (((size_t)row) << 10));
    v4f*       __restrict__ orow = (v4f*)      (Cout + (((size_t)row) << 10));
    const v4f* __restrict__ kr   = (const v4f*)(ksc  + (size_t)b * Hd);
    const v4f* __restrict__ qr   = (const v4f*)(qlin + (size_t)b * Hd);

    float acc = 0.0f;
#pragma unroll
    for (int it = 0; it < 8; ++it) {
        const int j = it * 32 + lane;         // v4f index, 256 per row
        v4f c  = __builtin_nontemporal_load(&crow[j]);
        v4f kv = kr[j];
        v4f qv = qr[j];
        v4f ct = fs * c + cs * kv;
        __builtin_nontemporal_store(ct, &orow[j]);
        acc += ct.x * qv.x + ct.y * qv.y + ct.z * qv.z + ct.w * qv.w;
    }
#pragma unroll
    for (int off = 16; off > 0; off >>= 1)
        acc += __shfl_down(acc, off, 32);
    if (lane == 0)
        hout[row] = o_t[row] * acc / divv[b];
}

// ---------------------------------------------------------------------------
extern "C" void kernel_launch(void* const* d_in, const int* in_sizes, int n_in,
                              void* d_out, int out_size, void* d_ws, size_t ws_size,
                              hipStream_t stream)
{
    (void)in_sizes; (void)n_in; (void)out_size; (void)ws_size;

    const float* x      = (const float*)d_in[0];
    const float* C      = (const float*)d_in[1];
    const float* n_in_p = (const float*)d_in[2];
    const float* m_in_p = (const float*)d_in[3];

    GemmArgs ga;
    for (int g = 0; g < 6; ++g) {
        ga.W[g]    = (const float*)d_in[4 + 2 * g];
        ga.bias[g] = (const float*)d_in[5 + 2 * g];
    }

    float* out  = (float*)d_out;
    float* h_t  = out;                               // (B,H)
    float* C_t  = out + BH;                          // (B,H,H)
    float* n_t  = out + BH + (size_t)BH * Hd;        // (B,H)
    float* m_t  = n_t + BH;                          // (B,H)

    float* ws   = (float*)d_ws;
    float* lin  = ws;                                // 6*B*H
    float* f_t  = ws + (size_t)6 * BH;
    float* o_t  = ws + (size_t)7 * BH;
    float* ipv  = ws + (size_t)8 * BH;               // i' * v
    float* ksc  = ws + (size_t)9 * BH;               // k / 32
    float* divv = ws + (size_t)10 * BH;              // (B,)
    float* qlin = lin + (size_t)3 * BH;              // q pre-activation

    // 1) six gate GEMMs (WMMA f32 16x16x4)
    gemm_gates_kernel<<<dim3(16, 6), 128, 0, stream>>>(x, ga, lin);

    // 2) pointwise gates + divisor reduction
    gates_kernel<<<Bsz, 256, 0, stream>>>(lin, n_in_p, m_in_p,
                                          f_t, o_t, ipv, ksc, divv, n_t, m_t);

    // 3) fused fast-weight update + mat-vec + output gate
    stream_kernel<<<(Bsz * Hd) / 8, 256, 0, stream>>>(C, f_t, o_t, ipv, ksc,
                                                      qlin, divv, C_t, h_t);
}